// BidirectionalRNN_17059610099717
// MI455X (gfx1250) — compile-verified
//
#include <hip/hip_runtime.h>

// Problem constants (reference: N=64, T=2048, D=1024, H=1024, fp32)
#define RNN_N 64
#define RNN_T 2048
#define RNN_D 1024
#define RNN_H 1024
#define NBLOCKS 256u   // (H/16) * (N/16) = 64 * 4 output tiles, one per block

typedef float v2f __attribute__((ext_vector_type(2)));
typedef float v8f __attribute__((ext_vector_type(8)));

// Zero-init h_0 (64*1024 floats).
__global__ void __launch_bounds__(1024) rnn_zero_kernel(float* __restrict__ h) {
    h[(size_t)blockIdx.x * 1024 + threadIdx.x] = 0.0f;
}

// Zero-init the per-step grid-barrier counters (RNN_T u32's). Must run every
// call: counters left behind by a previous call would break the barrier.
__global__ void __launch_bounds__(1024) rnn_zero_cnt_kernel(unsigned* __restrict__ c) {
    c[(size_t)blockIdx.x * 1024 + threadIdx.x] = 0u;
}

// Persistent RNN scan: one launch runs all T=2048 steps.
//   h_t = x_t @ Wxh + h_{t-1} @ Whh
//
// Block = 256 threads = 8 waves, owning ONE 16x16 output tile of h.
// Combined K = D + H = 2048 split 8 ways: wave w handles K in [256w, 256w+256).
// Waves 0..3 -> x_t @ Wxh slice, waves 4..7 -> h @ Whh slice.
// Per step: 64x V_WMMA_F32_16X16X4_F32 per wave, LDS reduction of the 8
// partials, tile store, then a device-wide barrier (per-step arrival counter)
// before the next step reads the new h. Weights stay resident in L2 for the
// whole scan; only x streams from HBM (prefetched one step ahead).
__global__ void __launch_bounds__(256) rnn_persistent_kernel(
    const float* __restrict__ x,    // (N, T, D)
    const float* __restrict__ Wxh,  // D x H row-major
    const float* __restrict__ Whh,  // H x H row-major
    float* __restrict__ hbuf0,      // N x H ping (holds h_0 = 0 on entry)
    float* __restrict__ hbuf1,      // N x H pong
    float* __restrict__ out,        // N x H final output
    unsigned* __restrict__ cnt)     // RNN_T arrival counters (zeroed on entry)
{
    __shared__ float red[8 * 256];  // 8 waves x (8 VGPRs x 32 lanes) partials

    const int tid   = threadIdx.x;
    const int wave  = tid >> 5;               // 0..7
    const int lane  = tid & 31;               // 0..31
    const int half  = lane >> 4;              // 0 or 1 (half-wave)
    const int ml    = lane & 15;              // row (A) / col (B) within tile
    const int bid   = blockIdx.x;             // 0..255
    const int nTile = (bid & 63) << 4;        // hidden-col tile base
    const int mTile = (bid >> 6) << 4;        // batch-row tile base

    // This wave's 256-wide slice of the combined K=2048.
    const int  kBase  = wave << 8;
    const bool phase2 = (kBase >= RNN_D);     // h @ Whh half
    const int  k0     = phase2 ? (kBase - RNN_D) : kBase;

    // B fragment base column: constant across all timesteps.
    const float* Bcol = (phase2 ? Whh : Wxh) + (size_t)k0 * RNN_H + nTile + ml;

    // Phase-1 A row walks x forward by D floats per step.
    const float* AxRow =
        x + (size_t)(mTile + ml) * (size_t)(RNN_T * RNN_D) + k0;
    // Phase-2 A row offset into whichever h buffer is "previous".
    const size_t hOff = (size_t)(mTile + ml) * RNN_H + k0;

    // Output element (C/D layout): VGPR i, lane L -> row i + 8*(L>>4), col L&15.
    const int oi   = tid >> 5;
    const int oL   = tid & 31;
    const size_t outIdx =
        (size_t)(mTile + oi + 8 * (oL >> 4)) * RNN_H + nTile + (oL & 15);

    for (int t = 0; t < RNN_T; ++t) {
        const float* Arow =
            phase2 ? ((t & 1) ? hbuf1 : hbuf0) + hOff : AxRow;

        v8f acc0 = {};
        v8f acc1 = {};

        #pragma unroll 4
        for (int k = 0; k < 256; k += 8) {
            const int ka = k + 2 * half;
            v2f a0 = *(const v2f*)(Arow + ka);
            v2f b0;
            b0.x = Bcol[(size_t)(ka    ) * RNN_H];
            b0.y = Bcol[(size_t)(ka + 1) * RNN_H];
            acc0 = __builtin_amdgcn_wmma_f32_16x16x4_f32(
                false, a0, false, b0, (short)0, acc0, false, false);

            const int kb = k + 4 + 2 * half;
            v2f a1 = *(const v2f*)(Arow + kb);
            v2f b1;
            b1.x = Bcol[(size_t)(kb    ) * RNN_H];
            b1.y = Bcol[(size_t)(kb + 1) * RNN_H];
            acc1 = __builtin_amdgcn_wmma_f32_16x16x4_f32(
                false, a1, false, b1, (short)0, acc1, false, false);
        }

        if (!phase2) {
            // Warm GL2 with next step's x slice (speculative prefetch; an OOB
            // address at the final step is silently dropped per the ISA).
            __builtin_prefetch(AxRow + RNN_D, 0, 0);
            AxRow += RNN_D;
        }

        v8f acc = acc0 + acc1;

        // Deposit this wave's partial tile into LDS.
        #pragma unroll
        for (int i = 0; i < 8; ++i) {
            red[wave * 256 + i * 32 + lane] = acc[i];
        }
        __syncthreads();

        // 256 threads reduce the 256 tile elements across the 8 wave partials.
        float s = 0.0f;
        #pragma unroll
        for (int w = 0; w < 8; ++w) {
            s += red[w * 256 + tid];
        }

        float* hout = (t == RNN_T - 1) ? out : ((t & 1) ? hbuf0 : hbuf1);
        hout[outIdx] = s;

        if (t < RNN_T - 1) {
            // Device-wide barrier: no block may start step t+1 (which reads
            // the h just written) until every block finished step t.
            __threadfence();            // release this thread's h stores (agent)
            __syncthreads();            // all threads' fences done; red reusable
            if (tid == 0) {
                __hip_atomic_fetch_add(&cnt[t], 1u, __ATOMIC_RELEASE,
                                       __HIP_MEMORY_SCOPE_AGENT);
                // Agent-scope acquire invalidates the WGP-level cache shared
                // by this whole block, so one waiter suffices.
                while (__hip_atomic_load(&cnt[t], __ATOMIC_ACQUIRE,
                                         __HIP_MEMORY_SCOPE_AGENT) < NBLOCKS) {
                    __builtin_amdgcn_s_sleep(1);
                }
            }
            __syncthreads();
        }
    }
}

extern "C" void kernel_launch(void* const* d_in, const int* in_sizes, int n_in,
                              void* d_out, int out_size, void* d_ws, size_t ws_size,
                              hipStream_t stream) {
    (void)in_sizes; (void)n_in; (void)out_size; (void)ws_size;

    const float* x   = (const float*)d_in[0];  // (N, T, D)
    const float* Wxh = (const float*)d_in[1];  // (D, H)
    const float* Whh = (const float*)d_in[2];  // (H, H)
    float* out = (float*)d_out;                // (N, H)

    // Workspace layout: h ping (256KB) | h pong (256KB) | step counters (8KB).
    float*    hbuf0 = (float*)d_ws;
    float*    hbuf1 = hbuf0 + (size_t)RNN_N * RNN_H;
    unsigned* cnt   = (unsigned*)(hbuf1 + (size_t)RNN_N * RNN_H);

    // Re-initialize per call (workspace is poisoned / left dirty by prior calls).
    rnn_zero_kernel<<<RNN_N, 1024, 0, stream>>>(hbuf0);
    rnn_zero_cnt_kernel<<<RNN_T / 1024, 1024, 0, stream>>>(cnt);

    // One persistent launch runs the whole 2048-step scan. 256 blocks of 256
    // threads (8 KB LDS, small VGPR footprint) co-reside trivially on an
    // MI455X-class part, which the software grid barrier requires.
    rnn_persistent_kernel<<<NBLOCKS, 256, 0, stream>>>(
        x, Wxh, Whh, hbuf0, hbuf1, out, cnt);
}